// GCN_22084721836225
// MI455X (gfx1250) — compile-verified
//
#include <hip/hip_runtime.h>
#include <hip/hip_bf16.h>

#define N_NODES 200000
#define N_EDGES 1000000
#define N_GRAPHS 4096
#define F_IN 11
#define H 64

typedef float v2f __attribute__((ext_vector_type(2)));
typedef float v8f __attribute__((ext_vector_type(8)));

__device__ __forceinline__ void atomicAddF(float* p, float v) {
    // hardware global_atomic_add_f32 (no CAS loop)
    unsafeAtomicAdd(p, v);
}

// ---------------- degree / norm ----------------

__global__ void k_init_deg(float* deg, int n) {
    int i = blockIdx.x * blockDim.x + threadIdx.x;
    if (i < n) deg[i] = 1.0f;   // self-loop
}

__global__ void k_deg_edges(const int* __restrict__ dst, float* deg, int e) {
    int i = blockIdx.x * blockDim.x + threadIdx.x;
    if (i < e) atomicAddF(&deg[dst[i]], 1.0f);
}

__global__ void k_rsqrt(float* deg, int n) {
    int i = blockIdx.x * blockDim.x + threadIdx.x;
    if (i < n) deg[i] = rsqrtf(fmaxf(deg[i], 1.0f));   // deg >= 1 always
}

// ---------------- WMMA GEMM: Out[N x 64] = A[N x KV] @ W[KV x 64] ----------------
// wave32: one wave computes a 16-row x 64-col strip via 4 accumulators of
// v_wmma_f32_16x16x4_f32. W is staged into LDS once per block, pair-interleaved
// so each B fragment is a single 8-byte ds_load. K-loop is fully unrolled at
// compile time; the only residual selects are the K=11 tail cndmasks.
//
// KV = valid K, AS = row stride of A, KP = padded K (multiple of 4).

template <int KV, int AS, int KP>
__global__ void k_gemm_wmma(const float* __restrict__ A, const float* __restrict__ W,
                            float* __restrict__ Out, int n) {
    // pair-interleaved weights: Wlds[(k/2)*H*2 + c*2 + (k&1)] = W[k][c] (0-padded)
    __shared__ float Wlds[(KP / 2) * H * 2];
    for (int idx = threadIdx.x; idx < KP * H; idx += blockDim.x) {
        int k = idx / H, c = idx - k * H;
        float v = (k < KV) ? W[k * H + c] : 0.0f;
        Wlds[(k >> 1) * (H * 2) + c * 2 + (k & 1)] = v;
    }
    __syncthreads();

    const int lane  = threadIdx.x & 31;
    const int strip = blockIdx.x * (blockDim.x >> 5) + (threadIdx.x >> 5);
    const int row0  = strip * 16;
    if (row0 >= n) return;

    const int mrow = lane & 15;   // M (A) / N (B,D) index covered by this lane
    const int hi   = lane >> 4;   // 0: K pair {0,1}; 1: K pair {2,3}
    const float* __restrict__ arow = A + (size_t)(row0 + mrow) * AS;

    v8f acc[4];
    #pragma unroll
    for (int t = 0; t < 4; ++t) acc[t] = (v8f){0, 0, 0, 0, 0, 0, 0, 0};

    #pragma unroll
    for (int k0 = 0; k0 < KP; k0 += 4) {
        const int ka = k0 + hi * 2;
        v2f a;
        if (KV == KP && (AS & 1) == 0) {
            // contiguous, 8B-aligned pair load (global_load_b64)
            a = *reinterpret_cast<const v2f*>(arow + ka);
        } else {
            // branch-free tail handling: clamped loads + value selects
            const int kc0 = (ka     < KV) ? ka     : (KV - 1);
            const int kc1 = (ka + 1 < KV) ? ka + 1 : (KV - 1);
            float ax = arow[kc0];
            float ay = arow[kc1];
            a.x = (ka     < KV) ? ax : 0.0f;
            a.y = (ka + 1 < KV) ? ay : 0.0f;
        }

        const int pbase = ((k0 >> 1) + hi) * (H * 2);
        #pragma unroll
        for (int t = 0; t < 4; ++t) {
            const int c = t * 16 + mrow;
            v2f b = *reinterpret_cast<const v2f*>(&Wlds[pbase + c * 2]);  // ds_load_b64
            acc[t] = __builtin_amdgcn_wmma_f32_16x16x4_f32(
                /*neg_a=*/false, a, /*neg_b=*/false, b,
                /*c_mod=*/(short)0, acc[t], /*reuse_a=*/false, /*reuse_b=*/false);
        }
    }

    // D layout: lanes 0-15 hold M=j (VGPR j), lanes 16-31 hold M=8+j; N = lane&15
    const int mbase = (lane < 16) ? 0 : 8;
    #pragma unroll
    for (int t = 0; t < 4; ++t) {
        const int col = t * 16 + mrow;
        #pragma unroll
        for (int j = 0; j < 8; ++j) {
            Out[(row0 + mbase + j) * H + col] = acc[t][j];
        }
    }
}

// ---------------- aggregation ----------------

// agg[i] = h[i] * dis[i]^2   (self-loop message; also zero-initializes agg)
__global__ void k_selfloop_init(const float* __restrict__ h, const float* __restrict__ dis,
                                float* __restrict__ agg, int n) {
    int t = blockIdx.x * blockDim.x + threadIdx.x;
    int node = t >> 4;
    if (node >= n) return;
    int f4 = (t & 15) * 4;
    float w = dis[node]; w *= w;
    float4 hv = *reinterpret_cast<const float4*>(h + node * H + f4);
    float4 r = make_float4(hv.x * w, hv.y * w, hv.z * w, hv.w * w);
    *reinterpret_cast<float4*>(agg + node * H + f4) = r;
}

// agg[dst] += h[src] * dis[src]*dis[dst]   (16 threads per edge, float4 gather)
__global__ void k_scatter_edges(const int* __restrict__ src, const int* __restrict__ dst,
                                const float* __restrict__ dis, const float* __restrict__ h,
                                float* __restrict__ agg, int e) {
    int t = blockIdx.x * blockDim.x + threadIdx.x;
    int edge = t >> 4;
    if (edge >= e) return;
    int f4 = (t & 15) * 4;
    int s = src[edge];
    int d = dst[edge];
    float w = dis[s] * dis[d];
    float4 hv = *reinterpret_cast<const float4*>(h + s * H + f4);
    float* o = agg + d * H + f4;
    atomicAddF(o + 0, hv.x * w);
    atomicAddF(o + 1, hv.y * w);
    atomicAddF(o + 2, hv.z * w);
    atomicAddF(o + 3, hv.w * w);
}

__global__ void k_bias_act(float* __restrict__ agg, const float* __restrict__ b,
                           int n, int relu) {
    int t = blockIdx.x * blockDim.x + threadIdx.x;
    if (t >= n * H) return;
    float v = agg[t] + b[t & (H - 1)];
    agg[t] = relu ? fmaxf(v, 0.0f) : v;
}

// ---------------- pooling + final linear ----------------

__global__ void k_zero(float* p, int n) {
    int i = blockIdx.x * blockDim.x + threadIdx.x;
    if (i < n) p[i] = 0.0f;
}

__global__ void k_pool_acc(const float* __restrict__ h, const int* __restrict__ batch,
                           float* __restrict__ sums, float* __restrict__ cnts, int n) {
    int t = blockIdx.x * blockDim.x + threadIdx.x;
    int node = t >> 4;
    if (node >= n) return;
    int f4 = (t & 15) * 4;
    int g = batch[node];
    float4 hv = *reinterpret_cast<const float4*>(h + node * H + f4);
    float* o = sums + g * H + f4;
    atomicAddF(o + 0, hv.x);
    atomicAddF(o + 1, hv.y);
    atomicAddF(o + 2, hv.z);
    atomicAddF(o + 3, hv.w);
    if ((t & 15) == 0) atomicAddF(&cnts[g], 1.0f);
}

__global__ void k_pool_linear(const float* __restrict__ sums, const float* __restrict__ cnts,
                              const float* __restrict__ lin_w, const float* __restrict__ lin_b,
                              float* __restrict__ out, int g_total) {
    int g = blockIdx.x * blockDim.x + threadIdx.x;
    if (g >= g_total) return;
    float inv = 1.0f / fmaxf(cnts[g], 1.0f);
    float acc = 0.0f;
    #pragma unroll 8
    for (int f = 0; f < H; ++f) acc += sums[g * H + f] * lin_w[f];
    out[g] = acc * inv + lin_b[0];
}

// ---------------- host orchestration ----------------

extern "C" void kernel_launch(void* const* d_in, const int* in_sizes, int n_in,
                              void* d_out, int out_size, void* d_ws, size_t ws_size,
                              hipStream_t stream) {
    const float* x     = (const float*)d_in[0];
    const int*   eidx  = (const int*)  d_in[1];
    const int*   batch = (const int*)  d_in[2];
    const float* W1    = (const float*)d_in[3];
    const float* b1    = (const float*)d_in[4];
    const float* W2    = (const float*)d_in[5];
    const float* b2    = (const float*)d_in[6];
    const float* W3    = (const float*)d_in[7];
    const float* b3    = (const float*)d_in[8];
    const float* lin_w = (const float*)d_in[9];
    const float* lin_b = (const float*)d_in[10];
    float* out = (float*)d_out;

    const int n = in_sizes[0] / F_IN;      // 200000
    const int e = in_sizes[1] / 2;         // 1000000
    const int* src = eidx;                 // edge_index[0]
    const int* dst = eidx + e;             // edge_index[1]

    // workspace carve-up (floats)
    float* ws   = (float*)d_ws;
    float* dis  = ws;                          // n
    float* bufA = dis  + n;                    // n*H  (gemm output h)
    float* bufB = bufA + (size_t)n * H;        // n*H  (aggregate / layer output)
    float* sums = bufB + (size_t)n * H;        // N_GRAPHS*H
    float* cnts = sums + (size_t)N_GRAPHS * H; // N_GRAPHS

    const int TB = 256;
    dim3 blk(TB);
    auto blocks = [](long long work, int tb) { return dim3((unsigned)((work + tb - 1) / tb)); };

    // ---- degree & symmetric norm ----
    k_init_deg  <<<blocks(n, TB), blk, 0, stream>>>(dis, n);
    k_deg_edges <<<blocks(e, TB), blk, 0, stream>>>(dst, dis, e);
    k_rsqrt     <<<blocks(n, TB), blk, 0, stream>>>(dis, n);

    const int strips       = (n + 15) / 16;    // 12500 waves (one 16-row strip each)
    const dim3 gemm_grid   = blocks((long long)strips * 32, TB);
    const long long nodef4 = (long long)n * 16;
    const long long edgef4 = (long long)e * 16;

    // ---- layer 1: in = x (K=11, row stride 11, padded to 12) ----
    k_gemm_wmma<F_IN, F_IN, 12><<<gemm_grid, blk, 0, stream>>>(x, W1, bufA, n);
    k_selfloop_init <<<blocks(nodef4, TB), blk, 0, stream>>>(bufA, dis, bufB, n);
    k_scatter_edges <<<blocks(edgef4, TB), blk, 0, stream>>>(src, dst, dis, bufA, bufB, e);
    k_bias_act      <<<blocks((long long)n * H, TB), blk, 0, stream>>>(bufB, b1, n, 1);

    // ---- layer 2: in = bufB (K=64) ----
    k_gemm_wmma<H, H, H><<<gemm_grid, blk, 0, stream>>>(bufB, W2, bufA, n);
    k_selfloop_init <<<blocks(nodef4, TB), blk, 0, stream>>>(bufA, dis, bufB, n);
    k_scatter_edges <<<blocks(edgef4, TB), blk, 0, stream>>>(src, dst, dis, bufA, bufB, e);
    k_bias_act      <<<blocks((long long)n * H, TB), blk, 0, stream>>>(bufB, b2, n, 1);

    // ---- layer 3: in = bufB (K=64), no relu ----
    k_gemm_wmma<H, H, H><<<gemm_grid, blk, 0, stream>>>(bufB, W3, bufA, n);
    k_selfloop_init <<<blocks(nodef4, TB), blk, 0, stream>>>(bufA, dis, bufB, n);
    k_scatter_edges <<<blocks(edgef4, TB), blk, 0, stream>>>(src, dst, dis, bufA, bufB, e);
    k_bias_act      <<<blocks((long long)n * H, TB), blk, 0, stream>>>(bufB, b3, n, 0);

    // ---- global mean pool + linear ----
    k_zero       <<<blocks((long long)N_GRAPHS * H, TB), blk, 0, stream>>>(sums, N_GRAPHS * H);
    k_zero       <<<blocks(N_GRAPHS, TB), blk, 0, stream>>>(cnts, N_GRAPHS);
    k_pool_acc   <<<blocks(nodef4, TB), blk, 0, stream>>>(bufB, batch, sums, cnts, n);
    k_pool_linear<<<blocks(N_GRAPHS, TB), blk, 0, stream>>>(sums, cnts, lin_w, lin_b, out, N_GRAPHS);
}